// GGNN_16063177687060
// MI455X (gfx1250) — compile-verified
//
#include <hip/hip_runtime.h>
#include <math.h>

// ---------------------------------------------------------------------------
// GCN (16 layers, D=50) on MI455X / gfx1250.
// Memory/atomic-bound on the edge scatter; GEMMs are fp32 via
// V_WMMA_F32_16X16X4_F32 (exact fp32 math). ReLU is fused into the next
// layer's A-fragment load, and the self-loop+bias agg-init is fused into the
// GEMM store, so each layer is just: gemm(+fusions) -> edge scatter.
// ---------------------------------------------------------------------------

typedef __attribute__((ext_vector_type(2))) float v2f;
typedef __attribute__((ext_vector_type(8))) float v8f;

#define DD 50   // true feature dim
#define DP 64   // padded row stride (multiple of 16)
#define DK 52   // padded K for WMMA (multiple of 4, >= 50)

// h[n][j] = x[n][j] for j<50, else 0 (rows n>=N zeroed too)
__global__ void k_pad_x(const float* __restrict__ x, float* __restrict__ h,
                        int N, int Np) {
    int i = blockIdx.x * blockDim.x + threadIdx.x;
    if (i >= Np * DP) return;
    int n = i >> 6, j = i & 63;
    h[i] = (n < N && j < DD) ? x[n * DD + j] : 0.0f;
}

__global__ void k_deg(const int* __restrict__ col, float* __restrict__ deg, int E) {
    int e = blockIdx.x * blockDim.x + threadIdx.x;
    if (e < E) atomicAdd(&deg[col[e]], 1.0f);
}

// dis[n] = rsqrt(deg[n] + 1)   (in place over padded range; padded rows -> 1)
__global__ void k_dis(float* __restrict__ dis, int Np) {
    int n = blockIdx.x * blockDim.x + threadIdx.x;
    if (n < Np) dis[n] = rsqrtf(dis[n] + 1.0f);
}

__global__ void k_norm(const int* __restrict__ row, const int* __restrict__ col,
                       const float* __restrict__ dis, float* __restrict__ norm, int E) {
    int e = blockIdx.x * blockDim.x + threadIdx.x;
    if (e < E) norm[e] = dis[row[e]] * dis[col[e]];
}

// Fused per-layer GEMM:
//   t   = relu_in ? relu(in) : in            (A fragments, each elem read once)
//   hw  = t @ Wpad                           (V_WMMA_F32_16X16X4_F32)
//   agg = hw * dis^2 + b                     (self-loop + bias; scatter adds later)
// in may alias agg (layer state); safe: each wave reads only its own 16 rows
// before overwriting them, and tiles partition rows across waves/blocks.
// WMMA fragment layouts (ISA 7.12.2):
//   A 16x4 f32 : lane(0-15)=rows M, VGPR0/1 = K0/K1; lanes 16-31 = K2/K3.
//   C/D 16x16  : lane&15 = N, lane>>4 selects M-base (0 or 8), VGPR v = M row.
//   B 4x16     : assumed symmetric to A (compile-only; not runtime-verifiable).
__global__ void k_gemm(const float* in, const float* __restrict__ W,
                       const float* __restrict__ bias, const float* __restrict__ dis,
                       float* __restrict__ hw, float* agg,
                       int ntiles, int relu_in) {
    __shared__ float Wt[DP * DK];   // transposed: Wt[n*52 + k] = W[k][n] (0-padded)
    __shared__ float bsh[DP];       // 0-padded bias
    for (int idx = threadIdx.x; idx < DP * DK; idx += blockDim.x) {
        int n = idx / DK, kk = idx - n * DK;
        Wt[idx] = (n < DD && kk < DD) ? W[kk * DD + n] : 0.0f;
    }
    if (threadIdx.x < DP) bsh[threadIdx.x] = (threadIdx.x < DD) ? bias[threadIdx.x] : 0.0f;
    __syncthreads();

    int wave = threadIdx.x >> 5;
    int lane = threadIdx.x & 31;
    int tile = blockIdx.x * 4 + wave;
    if (tile < ntiles) {                       // wave-uniform: EXEC stays all-ones
        int m  = lane & 15;
        int kh = lane >> 4;                    // K-half select (0/1)
        int nn = lane & 15;                    // N within a 16-wide tile
        const float* irow = in + (size_t)(tile * 16 + m) * DP;

        v8f acc0 = {}, acc1 = {}, acc2 = {}, acc3 = {};
        #pragma unroll
        for (int k = 0; k < DK; k += 4) {
            int k0 = k + 2 * kh;
            v2f a = *(const v2f*)(irow + k0);
            if (relu_in) { a.x = fmaxf(a.x, 0.0f); a.y = fmaxf(a.y, 0.0f); }
            v2f b0 = *(const v2f*)(Wt + (nn      ) * DK + k0);
            v2f b1 = *(const v2f*)(Wt + (nn + 16 ) * DK + k0);
            v2f b2 = *(const v2f*)(Wt + (nn + 32 ) * DK + k0);
            v2f b3 = *(const v2f*)(Wt + (nn + 48 ) * DK + k0);
            acc0 = __builtin_amdgcn_wmma_f32_16x16x4_f32(false, a, false, b0, (short)0, acc0, false, false);
            acc1 = __builtin_amdgcn_wmma_f32_16x16x4_f32(false, a, false, b1, (short)0, acc1, false, false);
            acc2 = __builtin_amdgcn_wmma_f32_16x16x4_f32(false, a, false, b2, (short)0, acc2, false, false);
            acc3 = __builtin_amdgcn_wmma_f32_16x16x4_f32(false, a, false, b3, (short)0, acc3, false, false);
        }

        float bb0 = bsh[nn], bb1 = bsh[nn + 16], bb2 = bsh[nn + 32], bb3 = bsh[nn + 48];
        int rbase = tile * 16 + 8 * kh;        // C/D: lane group selects M+8
        #pragma unroll
        for (int v = 0; v < 8; ++v) {
            int r = rbase + v;
            float sn = dis[r]; sn *= sn;       // 1/deg self-loop norm
            float* dh = hw  + (size_t)r * DP + nn;
            float* da = agg + (size_t)r * DP + nn;
            dh[0]  = acc0[v];  da[0]  = fmaf(acc0[v], sn, bb0);  // pad cols: 0*sn+0 = 0
            dh[16] = acc1[v];  da[16] = fmaf(acc1[v], sn, bb1);
            dh[32] = acc2[v];  da[32] = fmaf(acc2[v], sn, bb2);
            dh[48] = acc3[v];  da[48] = fmaf(acc3[v], sn, bb3);
        }
    }
}

// agg[col[e]][j] += hw[row[e]][j] * norm[e]   (one thread per (edge, j))
__global__ void k_scatter(const float* __restrict__ hw, const int* __restrict__ row,
                          const int* __restrict__ col, const float* __restrict__ norm,
                          float* __restrict__ agg, int E) {
    long long tid = (long long)blockIdx.x * blockDim.x + threadIdx.x;
    if (tid >= (long long)E * DP) return;
    int e = (int)(tid >> 6);
    int j = (int)(tid & 63);
    if (j >= DD) return;
    int r = row[e], c = col[e];
    atomicAdd(&agg[(size_t)c * DP + j], hw[(size_t)r * DP + j] * norm[e]);
}

// pooled[g][j] += relu(agg[n][j]) ; counts[g] += 1 (lane j==50 counts per node)
__global__ void k_pool(const float* __restrict__ agg, const int* __restrict__ batch,
                       float* __restrict__ pooled, float* __restrict__ counts, int N) {
    int i = blockIdx.x * blockDim.x + threadIdx.x;
    if (i >= N * DP) return;
    int n = i >> 6, j = i & 63;
    int g = batch[n];
    if (j < DD)       atomicAdd(&pooled[g * DD + j], fmaxf(agg[i], 0.0f));
    else if (j == DD) atomicAdd(&counts[g], 1.0f);
}

// out[g] = sigmoid(relu(mean_pooled @ W_h + b_h) @ W_f + b_f); one block per graph
__global__ void k_head(const float* __restrict__ pooled, const float* __restrict__ counts,
                       const float* __restrict__ W_h, const float* __restrict__ b_h,
                       const float* __restrict__ W_f, const float* __restrict__ b_f,
                       float* __restrict__ out) {
    __shared__ float h2s[DP];
    __shared__ float mean[DP];
    int g = blockIdx.x;
    int j = threadIdx.x;
    float inv = 1.0f / fmaxf(counts[g], 1.0f);
    mean[j] = (j < DD) ? pooled[g * DD + j] * inv : 0.0f;
    __syncthreads();
    if (j < DD) {
        float s = b_h[j];
        for (int k = 0; k < DD; ++k) s = fmaf(mean[k], W_h[k * DD + j], s);
        h2s[j] = fmaxf(s, 0.0f);
    } else {
        h2s[j] = 0.0f;
    }
    __syncthreads();
    if (j == 0) {
        float s = b_f[0];
        for (int k = 0; k < DD; ++k) s = fmaf(h2s[k], W_f[k], s);
        out[g] = 1.0f / (1.0f + expf(-s));
    }
}

static inline size_t align256(size_t v) { return (v + 255) & ~(size_t)255; }

extern "C" void kernel_launch(void* const* d_in, const int* in_sizes, int n_in,
                              void* d_out, int out_size, void* d_ws, size_t ws_size,
                              hipStream_t stream) {
    const float* x    = (const float*)d_in[0];
    const float* Ws   = (const float*)d_in[1];
    const float* bs   = (const float*)d_in[2];
    const float* W_h  = (const float*)d_in[3];
    const float* b_h  = (const float*)d_in[4];
    const float* W_f  = (const float*)d_in[5];
    const float* b_f  = (const float*)d_in[6];
    const int*   ei   = (const int*)d_in[7];
    const int*   bat  = (const int*)d_in[8];

    const int N = in_sizes[8];
    const int E = in_sizes[7] / 2;
    const int L = in_sizes[2] / DD;
    const int G = out_size;
    const int ntiles = (N + 15) / 16;
    const int Np = ntiles * 16;

    const int* row = ei;
    const int* col = ei + E;

    // workspace layout
    char* ws = (char*)d_ws;
    float* dis    = (float*)ws;  ws += align256((size_t)Np * sizeof(float));
    float* norm   = (float*)ws;  ws += align256((size_t)E  * sizeof(float));
    float* h0     = (float*)ws;  ws += align256((size_t)Np * DP * sizeof(float));
    float* hw     = (float*)ws;  ws += align256((size_t)Np * DP * sizeof(float));
    float* agg    = (float*)ws;  ws += align256((size_t)Np * DP * sizeof(float));
    float* pooled = (float*)ws;  ws += align256((size_t)G * DD * sizeof(float));
    float* counts = (float*)ws;

    const int B = 256;
    const int gNp = (Np * DP + B - 1) / B;

    // degree -> dis -> per-edge norm
    hipMemsetAsync(dis, 0, (size_t)Np * sizeof(float), stream);
    k_deg <<<(E + B - 1) / B, B, 0, stream>>>(col, dis, E);
    k_dis <<<(Np + B - 1) / B, B, 0, stream>>>(dis, Np);
    k_norm<<<(E + B - 1) / B, B, 0, stream>>>(row, col, dis, norm, E);

    // h0 = pad(x)
    k_pad_x<<<gNp, B, 0, stream>>>(x, h0, N, Np);

    // 16 GCN layers: fused gemm(+relu-in, +agg-init) then edge scatter
    long long stot = (long long)E * DP;
    int gScat = (int)((stot + B - 1) / B);
    for (int l = 0; l < L; ++l) {
        const float* in = (l == 0) ? h0 : agg;
        k_gemm   <<<(ntiles + 3) / 4, 128, 0, stream>>>(in, Ws + (size_t)l * DD * DD,
                                                        bs + (size_t)l * DD, dis,
                                                        hw, agg, ntiles, l > 0 ? 1 : 0);
        k_scatter<<<gScat, B, 0, stream>>>(hw, row, col, norm, agg, E);
    }

    // global mean pool (relu applied on read) + MLP head
    hipMemsetAsync(pooled, 0, (size_t)G * DD * sizeof(float), stream);
    hipMemsetAsync(counts, 0, (size_t)G * sizeof(float), stream);
    k_pool<<<(N * DP + B - 1) / B, B, 0, stream>>>(agg, bat, pooled, counts, N);
    k_head<<<G, DP, 0, stream>>>(pooled, counts, W_h, b_h, W_f, b_f, (float*)d_out);
}